// PositEpitopeEncoder_11570641895567
// MI455X (gfx1250) — compile-verified
//
#include <hip/hip_runtime.h>

// ---------------------------------------------------------------------------
// PositEpitopeEncoder for MI455X (gfx1250, wave32)
//
// out[i,:] = x[i,:] + (mask[i] ? pe[pos_i,:] : 0)
// pos_i = (inclusive masked-cumsum S[i]) - 1 - graphOffsets[batch[i]]
//
// Roofline: ~2.06 GB moved -> ~88 us @ 23.3 TB/s. Zero matmul content, so the
// relevant CDNA5 paths are data movement: async global->LDS double-buffered
// pipeline (ASYNCcnt) for the x stream, b128 NT loads/stores for the
// once-touched 2 GB, temporal caching for the reused 512 KB pe table.
// ---------------------------------------------------------------------------

typedef __attribute__((ext_vector_type(4))) float v4f;
typedef __attribute__((ext_vector_type(4))) int   v4i;

#define SCAN_TPB 256
#define SCAN_EPT 16
#define SCAN_EPB (SCAN_TPB * SCAN_EPT)   // 4096 elements per block

// ---------------- gfx1250 async global->LDS helpers ------------------------
#if defined(__gfx1250__) && __has_builtin(__builtin_amdgcn_global_load_async_to_lds_b128) && __has_builtin(__builtin_amdgcn_s_wait_asynccnt)
#define HAVE_ASYNC_BUILTIN 1
#else
#define HAVE_ASYNC_BUILTIN 0
#endif

// Builtin signature wants pointers to 128-bit int vectors in AS1 / AS3.
typedef __attribute__((address_space(1))) v4i* as1_v4i_p;
typedef __attribute__((address_space(3))) v4i* as3_v4i_p;

__device__ __forceinline__ void asyncCopyB128(const void* g, void* lds_generic) {
    // flat->LDS addrspacecast on AMDGPU == truncate to low 32 bits.
    unsigned lds32 = (unsigned)(unsigned long long)lds_generic;
#if HAVE_ASYNC_BUILTIN
    as1_v4i_p gp = (as1_v4i_p)(unsigned long long)g;
    as3_v4i_p lp = (as3_v4i_p)lds32;
    __builtin_amdgcn_global_load_async_to_lds_b128(gp, lp, 0, 0);
#else
    asm volatile("global_load_async_to_lds_b128 %0, %1, off"
                 :: "v"(lds32), "v"(g) : "memory");
#endif
}

__device__ __forceinline__ void asyncWait(int n) {  // n is a literal 0 or 1
#if HAVE_ASYNC_BUILTIN
    if (n == 0) __builtin_amdgcn_s_wait_asynccnt(0);
    else        __builtin_amdgcn_s_wait_asynccnt(1);
#else
    if (n == 0) asm volatile("s_wait_asynccnt 0x0" ::: "memory");
    else        asm volatile("s_wait_asynccnt 0x1" ::: "memory");
#endif
    asm volatile("" ::: "memory");  // keep LDS reads below the wait
}

// ---------------- Pass A: per-block mask sums -------------------------------
__global__ void maskBlockSum(const int* __restrict__ mask, int n,
                             int* __restrict__ blockSums) {
    int t = threadIdx.x;
    int base = blockIdx.x * SCAN_EPB + t * SCAN_EPT;
    bool full = (long long)(blockIdx.x + 1) * SCAN_EPB <= (long long)n;
    int s = 0;
    if (full) {
        const v4i* mp = (const v4i*)(mask + base);
#pragma unroll
        for (int k = 0; k < SCAN_EPT / 4; ++k) {
            v4i m = mp[k];
            s += (m.x != 0) + (m.y != 0) + (m.z != 0) + (m.w != 0);
        }
    } else {
#pragma unroll
        for (int k = 0; k < SCAN_EPT; ++k) {
            int idx = base + k;
            if (idx < n) s += (mask[idx] != 0);
        }
    }
    __shared__ int sh[SCAN_TPB];
    sh[t] = s;
    __syncthreads();
    for (int off = SCAN_TPB >> 1; off > 0; off >>= 1) {
        if (t < off) sh[t] += sh[t + off];
        __syncthreads();
    }
    if (t == 0) blockSums[blockIdx.x] = sh[0];
}

// ---------------- Pass B: serial exclusive scan of 245 block sums -----------
__global__ void scanBlockSums(const int* __restrict__ blockSums, int nb,
                              int* __restrict__ blockOffsets) {
    if (threadIdx.x == 0 && blockIdx.x == 0) {
        int acc = 0;
        for (int j = 0; j < nb; ++j) { blockOffsets[j] = acc; acc += blockSums[j]; }
    }
}

// ---------------- Pass C: full scan S[i] + per-graph offsets ----------------
__global__ void maskScanGraphOffsets(const int* __restrict__ mask,
                                     const int* __restrict__ batch,
                                     const int* __restrict__ blockOffsets,
                                     int n,
                                     int* __restrict__ S,
                                     int* __restrict__ graphOffsets) {
    int t = threadIdx.x;
    int base = blockIdx.x * SCAN_EPB + t * SCAN_EPT;
    bool full = (long long)(blockIdx.x + 1) * SCAN_EPB <= (long long)n;

    int m[SCAN_EPT];
    int b[SCAN_EPT];
    int run = 0;
    if (full) {
        const v4i* mp = (const v4i*)(mask + base);
        const v4i* bp = (const v4i*)(batch + base);
#pragma unroll
        for (int k = 0; k < SCAN_EPT / 4; ++k) {
            v4i mv = mp[k];
            v4i bv = bp[k];
            m[4 * k + 0] = (mv.x != 0); m[4 * k + 1] = (mv.y != 0);
            m[4 * k + 2] = (mv.z != 0); m[4 * k + 3] = (mv.w != 0);
            b[4 * k + 0] = bv.x; b[4 * k + 1] = bv.y;
            b[4 * k + 2] = bv.z; b[4 * k + 3] = bv.w;
            run += m[4 * k] + m[4 * k + 1] + m[4 * k + 2] + m[4 * k + 3];
        }
    } else {
#pragma unroll
        for (int k = 0; k < SCAN_EPT; ++k) {
            int idx = base + k;
            int mv = (idx < n) ? (mask[idx] != 0) : 0;
            m[k] = mv;
            b[k] = (idx < n) ? batch[idx] : -1;
            run += mv;
        }
    }

    __shared__ int sh[SCAN_TPB];
    sh[t] = run;
    __syncthreads();
    for (int off = 1; off < SCAN_TPB; off <<= 1) {   // Hillis-Steele inclusive
        int v = (t >= off) ? sh[t - off] : 0;
        __syncthreads();
        sh[t] += v;
        __syncthreads();
    }
    int acc = blockOffsets[blockIdx.x] + sh[t] - run;  // exclusive thread prefix

    int bprev = (base > 0 && base <= n) ? batch[base - 1] : -1;
#pragma unroll
    for (int k = 0; k < SCAN_EPT; ++k) {
        int idx = base + k;
        if (idx < n) {
            acc += m[k];
            S[idx] = acc;
            int bc = b[k];
            if (bc != bprev) graphOffsets[bc] = acc - m[k];
            bprev = bc;
        }
    }
}

// ---------------- Pass D: fold everything into posArr[i] --------------------
__global__ void computePos(const int* __restrict__ mask,
                           const int* __restrict__ batch,
                           const int* __restrict__ S,
                           const int* __restrict__ graphOffsets,
                           int* __restrict__ posArr, int n, int maxLen) {
    int i = blockIdx.x * blockDim.x + threadIdx.x;
    if (i >= n) return;
    int p = -1;
    if (mask[i] != 0) {
        p = S[i] - 1 - graphOffsets[batch[i]];
        p = p < 0 ? 0 : (p >= maxLen ? maxLen - 1 : p);
    }
    posArr[i] = p;
}

// ---------------- Pass E: 2 GB stream, async double-buffered ----------------
#define E_TPB 256            // 8 wave32 waves
#define E_TILE E_TPB         // 256 float4 per tile (4 KB)

__global__ void addPosEnc(const float* __restrict__ x,
                          const float* __restrict__ pe,
                          const int* __restrict__ posArr,
                          float* __restrict__ out,
                          int totalF4,          // 64M fits int32
                          int d4shift,          // log2(D/4) = 6
                          int nodesPerTile,     // E_TILE >> d4shift = 4
                          int tilesPerBlock) {
    __shared__ v4f buf[2][E_TPB];
    int t = threadIdx.x;
    int tile0 = blockIdx.x * tilesPerBlock;
    const v4f* xp4 = (const v4f*)x;
    const v4f* pe4 = (const v4f*)pe;
    v4f* out4 = (v4f*)out;

    // Per-thread lane decomposition is loop-invariant: hoist it.
    int nit = t >> d4shift;                 // node within tile
    int d4  = t & ((1 << d4shift) - 1);     // float4 column within node row

    // Prime the pipeline: issue tile 0 if fully in-bounds.
    if (tile0 * E_TILE + E_TILE <= totalF4)
        asyncCopyB128(xp4 + tile0 * E_TILE + t, &buf[0][t]);

    for (int j = 0; j < tilesPerBlock; ++j) {
        int tileIdx = tile0 + j;
        int f4base = tileIdx * E_TILE;
        if (f4base >= totalF4) return;            // uniform across block
        int f4 = f4base + t;
        bool fullTile = (f4base + E_TILE) <= totalF4;   // uniform

        v4f v;
        if (fullTile) {
            bool nextFull = (j + 1 < tilesPerBlock) &&
                            ((f4base + 2 * E_TILE) <= totalF4);
            if (nextFull) {
                asyncCopyB128(xp4 + (f4base + E_TILE) + t, &buf[(j + 1) & 1][t]);
                asyncWait(1);   // tile j complete; tile j+1 in flight
            } else {
                asyncWait(0);
            }
            v = buf[j & 1][t];                    // ds_load_b128 (own slot)
        } else if (f4 < totalF4) {
            v = __builtin_nontemporal_load(xp4 + f4);
        }

        if (f4 < totalF4) {
            int node = tileIdx * nodesPerTile + nit;
            int p = posArr[node];                 // uniform per 64 lanes
            if (p >= 0) v += pe4[(p << d4shift) + d4];
            __builtin_nontemporal_store(v, out4 + f4);
        }
    }
}

// ---------------------------------------------------------------------------
extern "C" void kernel_launch(void* const* d_in, const int* in_sizes, int n_in,
                              void* d_out, int out_size, void* d_ws, size_t ws_size,
                              hipStream_t stream) {
    const float* x     = (const float*)d_in[0];
    const float* pe    = (const float*)d_in[1];
    const int*   mask  = (const int*)d_in[2];
    const int*   batch = (const int*)d_in[3];

    int N  = in_sizes[2];               // 1,000,000
    int D  = in_sizes[0] / N;           // 256
    int L  = in_sizes[1] / D;           // 512
    int D4 = D / 4;                     // 64 (power of two)
    float* out = (float*)d_out;

    int d4shift = 0;
    while ((1 << d4shift) < D4) ++d4shift;   // 6

    // Workspace (ints): S[N], graphOffsets[N], posArr[N], blockSums/Offsets.
    int* S            = (int*)d_ws;
    int* graphOffsets = S + N;
    int* posArr       = graphOffsets + N;
    int  nbA          = (N + SCAN_EPB - 1) / SCAN_EPB;   // 245
    int* blockSums    = posArr + N;
    int* blockOffsets = blockSums + nbA;

    maskBlockSum<<<nbA, SCAN_TPB, 0, stream>>>(mask, N, blockSums);
    scanBlockSums<<<1, 32, 0, stream>>>(blockSums, nbA, blockOffsets);
    maskScanGraphOffsets<<<nbA, SCAN_TPB, 0, stream>>>(mask, batch, blockOffsets,
                                                       N, S, graphOffsets);
    computePos<<<(N + 255) / 256, 256, 0, stream>>>(mask, batch, S, graphOffsets,
                                                    posArr, N, L);

    int totalF4 = N * D4;                                // 64M float4
    int nodesPerTile = E_TILE >> d4shift;                // 4
    int tilesPerBlock = 8;
    int totalTiles = (totalF4 + E_TILE - 1) / E_TILE;    // 250,000
    int grid = (totalTiles + tilesPerBlock - 1) / tilesPerBlock;  // 31,250
    addPosEnc<<<grid, E_TPB, 0, stream>>>(x, pe, posArr, out,
                                          totalF4, d4shift, nodesPerTile,
                                          tilesPerBlock);
}